// WeightRegularizationLoss_50586124812489
// MI455X (gfx1250) — compile-verified
//
#include <hip/hip_runtime.h>
#include <math.h>

// ---------------------------------------------------------------------------
// WeightRegularizationLoss for MI455X (gfx1250, wave32)
//
// Bandwidth-bound streaming reduction: ~464 MB read once -> ~20us floor at
// 23.3 TB/s. Design: non-temporal b128 loads (read-once data, 464MB > 192MB
// L2 so caching is pure pollution), unroll-by-4 grid-stride loop (4 NT B128
// loads in flight per thread), and a WMMA-based cross-lane reduction
// (V_WMMA_F32_16X16X4_F32 with B = ones acts as a 32-lane adder tree).
// Fully deterministic: fixed-shape two-stage reductions, no float atomics.
// ---------------------------------------------------------------------------

typedef __attribute__((ext_vector_type(4))) float f4;
typedef __attribute__((ext_vector_type(2))) float v2f;
typedef __attribute__((ext_vector_type(8))) float v8f;

#define BLOCK       256    // fixed everywhere -> 8 waves, hardcoded reductions
#define NWAVES      (BLOCK / 32)
#define PART_STRIDE 4096   // max partial blocks per matrix
#define MAXB        4096
#define NMAT        7
#define NROWS       128
#define ROWLEN      2048

// --- wave32 sum via one WMMA -----------------------------------------------
// A (16x4 f32): lane m<16 holds A[m][0]=v, A[m][1]=0; lane m+16 holds
// A[m][2]=v, A[m][3]=0.  B = all ones (4x16).  Then D[m][n] = v_m + v_{m+16}.
// D layout: VGPR j -> lanes 0-15 hold r_j, lanes 16-31 hold r_{j+8}.
// Sum the 8 local D regs, then xor-16 swap-add => full 32-lane sum in every lane.
__device__ __forceinline__ float wave_reduce_wmma(float v) {
    v2f a; a.x = v;   a.y = 0.0f;
    v2f b; b.x = 1.0f; b.y = 1.0f;
    v8f c = {};
    c = __builtin_amdgcn_wmma_f32_16x16x4_f32(
            /*neg_a=*/false, a, /*neg_b=*/false, b,
            /*c_mod=*/(short)0, c, /*reuse_a=*/false, /*reuse_b=*/false);
    float s = ((c[0] + c[1]) + (c[2] + c[3])) + ((c[4] + c[5]) + (c[6] + c[7]));
    s += __shfl_xor(s, 16, 32);
    return s;
}

// Block-wide sum for BLOCK==256 (result valid in thread 0 only).
__device__ __forceinline__ float block_reduce(float v) {
    __shared__ float wsum[NWAVES];
    float w = wave_reduce_wmma(v);
    const int lane = threadIdx.x & 31;
    const int wid  = threadIdx.x >> 5;
    if (lane == 0) wsum[wid] = w;
    __syncthreads();
    float t = 0.0f;
    if (threadIdx.x == 0) {
#pragma unroll
        for (int i = 0; i < NWAVES; ++i) t += wsum[i];  // fixed order: deterministic
    }
    return t;
}

// --- stage 1: per-block partial sum of squares over one matrix --------------
__global__ void __launch_bounds__(BLOCK)
wrl_sumsq_partial(const float* __restrict__ p, long long n4,
                  float* __restrict__ out) {
    const f4* __restrict__ src = (const f4*)p;
    const long long stride = (long long)gridDim.x * BLOCK;
    const long long idx0   = (long long)blockIdx.x * BLOCK + threadIdx.x;

    float a0 = 0.0f, a1 = 0.0f, a2 = 0.0f, a3 = 0.0f;
    long long i = idx0;
    // main loop: 4 independent NT b128 loads in flight per thread
    for (; i + 3 * stride < n4; i += 4 * stride) {
        f4 v0 = __builtin_nontemporal_load(&src[i]);
        f4 v1 = __builtin_nontemporal_load(&src[i + stride]);
        f4 v2 = __builtin_nontemporal_load(&src[i + 2 * stride]);
        f4 v3 = __builtin_nontemporal_load(&src[i + 3 * stride]);
        a0 = fmaf(v0.x, v0.x, a0); a0 = fmaf(v0.y, v0.y, a0);
        a0 = fmaf(v0.z, v0.z, a0); a0 = fmaf(v0.w, v0.w, a0);
        a1 = fmaf(v1.x, v1.x, a1); a1 = fmaf(v1.y, v1.y, a1);
        a1 = fmaf(v1.z, v1.z, a1); a1 = fmaf(v1.w, v1.w, a1);
        a2 = fmaf(v2.x, v2.x, a2); a2 = fmaf(v2.y, v2.y, a2);
        a2 = fmaf(v2.z, v2.z, a2); a2 = fmaf(v2.w, v2.w, a2);
        a3 = fmaf(v3.x, v3.x, a3); a3 = fmaf(v3.y, v3.y, a3);
        a3 = fmaf(v3.z, v3.z, a3); a3 = fmaf(v3.w, v3.w, a3);
    }
    // tail
    for (; i < n4; i += stride) {
        f4 v = __builtin_nontemporal_load(&src[i]);
        a0 = fmaf(v.x, v.x, a0); a0 = fmaf(v.y, v.y, a0);
        a0 = fmaf(v.z, v.z, a0); a0 = fmaf(v.w, v.w, a0);
    }
    float acc = (a0 + a1) + (a2 + a3);
    // loops have reconverged here -> EXEC all-ones, safe for WMMA
    float t = block_reduce(acc);
    if (threadIdx.x == 0) out[blockIdx.x] = t;
}

// --- stage 1b: per-row L2 norm of biases (one block per row) ----------------
__global__ void __launch_bounds__(BLOCK)
wrl_bias_rownorm(const float* __restrict__ b, float* __restrict__ rows) {
    const f4* __restrict__ src = (const f4*)(b + (long long)blockIdx.x * ROWLEN);
    // ROWLEN/4 = 512 f4 over 256 threads -> exactly 2 per thread
    f4 v0 = __builtin_nontemporal_load(&src[threadIdx.x]);
    f4 v1 = __builtin_nontemporal_load(&src[threadIdx.x + BLOCK]);
    float a0 = 0.0f, a1 = 0.0f;
    a0 = fmaf(v0.x, v0.x, a0); a0 = fmaf(v0.y, v0.y, a0);
    a0 = fmaf(v0.z, v0.z, a0); a0 = fmaf(v0.w, v0.w, a0);
    a1 = fmaf(v1.x, v1.x, a1); a1 = fmaf(v1.y, v1.y, a1);
    a1 = fmaf(v1.z, v1.z, a1); a1 = fmaf(v1.w, v1.w, a1);
    float t = block_reduce(a0 + a1);
    if (threadIdx.x == 0) rows[blockIdx.x] = sqrtf(t);
}

// --- stage 2: reduce each matrix's block partials -> Frobenius norm ---------
struct Counts { int c[NMAT]; };

__global__ void __launch_bounds__(BLOCK)
wrl_reduce_norms(const float* __restrict__ ws, float* __restrict__ norms,
                 Counts cnt) {
    const int m = blockIdx.x;
    const int n = cnt.c[m];
    const float* __restrict__ part = ws + (long long)m * PART_STRIDE;
    float acc = 0.0f;
    for (int i = threadIdx.x; i < n; i += BLOCK) acc += part[i];
    float t = block_reduce(acc);
    if (threadIdx.x == 0) norms[m] = sqrtf(t);
}

// --- stage 3: combine 7 matrix norms + 128 row norms -> scalar --------------
__global__ void __launch_bounds__(BLOCK)
wrl_finalize(const float* __restrict__ rows, const float* __restrict__ norms,
             float* __restrict__ out) {
    const int t = threadIdx.x;
    float v = 0.0f;
    if (t < NROWS)              v = rows[t];
    else if (t < NROWS + NMAT)  v = norms[t - NROWS];
    // reconverged -> full EXEC for the WMMA reduce
    float s = block_reduce(v);
    if (t == 0) out[0] = 0.0001f * (s / (float)(NMAT + NROWS));
}

// ---------------------------------------------------------------------------
extern "C" void kernel_launch(void* const* d_in, const int* in_sizes, int n_in,
                              void* d_out, int out_size, void* d_ws, size_t ws_size,
                              hipStream_t stream) {
    (void)n_in; (void)out_size; (void)ws_size;

    const float* biases = (const float*)d_in[7];
    float* ws    = (float*)d_ws;
    float* rows  = ws + (long long)NMAT * PART_STRIDE;  // 128 row norms
    float* norms = rows + NROWS;                        // 7 matrix norms

    Counts cnt;
    for (int i = 0; i < NMAT; ++i) {
        const float* p = (const float*)d_in[i];
        long long n4 = (long long)in_sizes[i] >> 2;     // all sizes % 4 == 0
        int nb = (int)((n4 + BLOCK - 1) / BLOCK);
        if (nb > MAXB) nb = MAXB;
        cnt.c[i] = nb;
        wrl_sumsq_partial<<<nb, BLOCK, 0, stream>>>(p, n4,
                                                    ws + (long long)i * PART_STRIDE);
    }

    wrl_bias_rownorm<<<NROWS, BLOCK, 0, stream>>>(biases, rows);
    wrl_reduce_norms<<<NMAT, BLOCK, 0, stream>>>(ws, norms, cnt);
    wrl_finalize<<<1, BLOCK, 0, stream>>>(rows, norms, (float*)d_out);
}